// GSS_GNNLayer_1649267442177
// MI455X (gfx1250) — compile-verified
//
#include <hip/hip_runtime.h>
#include <cmath>

typedef __attribute__((ext_vector_type(2))) float v2f;
typedef __attribute__((ext_vector_type(8))) float v8f;

#define D_DIM 128

// ---------------------------------------------------------------------------
// Zero the accumulation workspace (atomics accumulate; must re-zero per call).
// ---------------------------------------------------------------------------
__global__ void zero_kernel(float* __restrict__ p, int n) {
  int i = blockIdx.x * blockDim.x + threadIdx.x;
  if (i < n) p[i] = 0.0f;
}

// ---------------------------------------------------------------------------
// SpMM 1:  y[row[e], :] += val[e] * x[col[e], :]
// One wave32 per edge; lane handles 2 contiguous columns per step (2 steps).
// 128B-coalesced float2 loads; native fp32 global atomics for the scatter.
// ---------------------------------------------------------------------------
__global__ void spmm1_kernel(const int* __restrict__ row,
                             const int* __restrict__ col,
                             const float* __restrict__ val,
                             const float* __restrict__ x,
                             float* __restrict__ y, int E) {
  int gid  = blockIdx.x * blockDim.x + threadIdx.x;
  int e    = gid >> 5;
  int lane = gid & 31;
  if (e >= E) return;
  int   r = row[e];
  int   c = col[e];
  float v = val[e];
  const float* xs = x + (size_t)c * D_DIM;
  float*       yd = y + (size_t)r * D_DIM;
#pragma unroll
  for (int j = 0; j < D_DIM; j += 64) {
    int jj = j + lane * 2;
    v2f xv = *(const v2f*)(xs + jj);
    unsafeAtomicAdd(&yd[jj + 0], v * xv.x);
    unsafeAtomicAdd(&yd[jj + 1], v * xv.y);
  }
}

// ---------------------------------------------------------------------------
// SpMM 2:  y[row[e], :] += val[e] * (Ax[col[e], :] * x[col[e], :])
// Fuses the elementwise Ax*X product into the gather (no extra pass/buffer).
// ---------------------------------------------------------------------------
__global__ void spmm2_kernel(const int* __restrict__ row,
                             const int* __restrict__ col,
                             const float* __restrict__ val,
                             const float* __restrict__ ax,
                             const float* __restrict__ x,
                             float* __restrict__ y, int E) {
  int gid  = blockIdx.x * blockDim.x + threadIdx.x;
  int e    = gid >> 5;
  int lane = gid & 31;
  if (e >= E) return;
  int   r = row[e];
  int   c = col[e];
  float v = val[e];
  const float* as = ax + (size_t)c * D_DIM;
  const float* xs = x  + (size_t)c * D_DIM;
  float*       yd = y  + (size_t)r * D_DIM;
#pragma unroll
  for (int j = 0; j < D_DIM; j += 64) {
    int jj = j + lane * 2;
    v2f av = *(const v2f*)(as + jj);
    v2f xv = *(const v2f*)(xs + jj);
    unsafeAtomicAdd(&yd[jj + 0], v * av.x * xv.x);
    unsafeAtomicAdd(&yd[jj + 1], v * av.y * xv.y);
  }
}

// ---------------------------------------------------------------------------
// Fused dual-GEMM + bias + ELU using V_WMMA_F32_16X16X4_F32 (pure fp32 path,
// matching the fp32 reference).  One wave per 16x16 output tile.
//
//   pre = Ax @ W1^T + Axx @ W2^T + (b1 + b2) ;  out = elu(pre)
//
// Operand layout per ISA 7.12.2 (fp32, wave32):
//   A 16x4 : lane&15 = M row; VGPR0 holds K = {0 | 2} by lane>>4, VGPR1 = {1|3}
//            -> per-lane float2 load from Ax[row][k + (lane>>4)*2]
//   B 4x16 : mirrored; B[k][n] = W[n][k] -> float2 load from W[n][k + (lane>>4)*2]
//   C 16x16: 8 VGPRs; lanes<16: M = i, lanes>=16: M = i+8, N = lane&15
// ---------------------------------------------------------------------------
__global__ void gemm_bias_elu_kernel(const float* __restrict__ Ax,
                                     const float* __restrict__ Axx,
                                     const float* __restrict__ W1,
                                     const float* __restrict__ b1,
                                     const float* __restrict__ W2,
                                     const float* __restrict__ b2,
                                     float* __restrict__ pre_out,
                                     float* __restrict__ act_out,
                                     int N) {
  const int D = D_DIM;
  int wid  = (blockIdx.x * blockDim.x + threadIdx.x) >> 5;
  int lane = threadIdx.x & 31;
  const int ntiles = D / 16;              // 8 column tiles
  int mtile = wid / ntiles;
  int ntile = wid - mtile * ntiles;
  if (mtile * 16 >= N) return;            // wave-uniform: EXEC stays all-ones

  int row0  = mtile * 16;
  int n0    = ntile * 16;
  int mrow  = lane & 15;                  // A: M row / B: N column
  int khalf = (lane >> 4) * 2;            // K sub-select {0,2}

  const float* arow  = Ax  + (size_t)(row0 + mrow) * D + khalf;
  const float* a2row = Axx + (size_t)(row0 + mrow) * D + khalf;
  const float* w1row = W1  + (size_t)(n0 + mrow) * D + khalf;   // B = W1^T
  const float* w2row = W2  + (size_t)(n0 + mrow) * D + khalf;   // B = W2^T

  v8f c = {};

  // GEMM 1: Ax @ W1^T  (32 x v_wmma_f32_16x16x4_f32)
#pragma unroll
  for (int k = 0; k < D; k += 4) {
    v2f a = *(const v2f*)(arow + k);
    v2f b = *(const v2f*)(w1row + k);
    c = __builtin_amdgcn_wmma_f32_16x16x4_f32(
        /*neg_a=*/false, a, /*neg_b=*/false, b,
        /*c_mod=*/(short)0, c, /*reuse_a=*/false, /*reuse_b=*/false);
  }

  // GEMM 2: Axx @ W2^T, accumulating into the same C tile
#pragma unroll
  for (int k = 0; k < D; k += 4) {
    v2f a = *(const v2f*)(a2row + k);
    v2f b = *(const v2f*)(w2row + k);
    c = __builtin_amdgcn_wmma_f32_16x16x4_f32(
        false, a, false, b, (short)0, c, false, false);
  }

  // Bias (per output column) + ELU, write both tuple outputs.
  int   colIdx = n0 + mrow;
  float bias   = b1[colIdx] + b2[colIdx];
  int   mbase  = row0 + (lane >> 4) * 8;
#pragma unroll
  for (int i = 0; i < 8; i++) {
    float p = c[i] + bias;
    float o = (p > 0.0f) ? p : expm1f(p);
    size_t idx = (size_t)(mbase + i) * D + colIdx;
    pre_out[idx] = p;
    act_out[idx] = o;
  }
}

// ---------------------------------------------------------------------------
// Launch
// ---------------------------------------------------------------------------
extern "C" void kernel_launch(void* const* d_in, const int* in_sizes, int n_in,
                              void* d_out, int out_size, void* d_ws, size_t ws_size,
                              hipStream_t stream) {
  const float* features = (const float*)d_in[0];
  const int*   adj_row  = (const int*)d_in[1];
  const int*   adj_col  = (const int*)d_in[2];
  const float* adj_val  = (const float*)d_in[3];
  const float* W1       = (const float*)d_in[4];
  const float* b1       = (const float*)d_in[5];
  const float* W2       = (const float*)d_in[6];
  const float* b2       = (const float*)d_in[7];

  const int D = D_DIM;
  const int N = in_sizes[0] / D;   // 40000
  const int E = in_sizes[1];       // 640000

  float* Ax  = (float*)d_ws;                // [N, D] accumulators
  float* Axx = Ax + (size_t)N * D;          // [N, D]
  float* pre_out = (float*)d_out;           // tuple element 0
  float* act_out = pre_out + (size_t)N * D; // tuple element 1

  // 1) zero the two accumulation buffers
  int ztot = 2 * N * D;
  zero_kernel<<<(ztot + 255) / 256, 256, 0, stream>>>(Ax, ztot);

  // 2) Ax = A @ X
  long long spmm_threads = (long long)E * 32;
  int spmm_blocks = (int)((spmm_threads + 255) / 256);
  spmm1_kernel<<<spmm_blocks, 256, 0, stream>>>(adj_row, adj_col, adj_val,
                                                features, Ax, E);

  // 3) Axx = A @ (Ax * X)
  spmm2_kernel<<<spmm_blocks, 256, 0, stream>>>(adj_row, adj_col, adj_val,
                                                Ax, features, Axx, E);

  // 4) pre = Ax@W1^T + Axx@W2^T + b1 + b2 ; out = elu(pre)
  int waves  = (N / 16) * (D / 16);         // 20000 tiles, one wave each
  int blocks = (waves * 32 + 255) / 256;    // 8 waves / 256-thread block
  gemm_bias_elu_kernel<<<blocks, 256, 0, stream>>>(Ax, Axx, W1, b1, W2, b2,
                                                   pre_out, act_out, N);
}